// Net_49203145343455
// MI455X (gfx1250) — compile-verified
//
#include <hip/hip_runtime.h>
#include <math.h>

// ---------------------------------------------------------------------------
// GraphSAGE 2-layer forward for MI455X (gfx1250, wave32, WMMA).
//
// v2: CSR pull-aggregation instead of 269M global f32 atomics:
//   deg count (int atomics) -> LDS exclusive scan -> bucket edges by dst,
//   then one wave per node accumulates its in-edges in registers (no atomics).
// GEMMs: persistent blocks, B-slice held in 64 VGPRs across row tiles,
//   A tile staged in LDS (stride 132 floats -> conflict-free ds_load_b64),
//   V_WMMA_F32_16X16X4_F32 inner loop, fused bias/mean/ReLU epilogues.
// Whole inter-kernel working set (~115MB) is resident in the 192MB L2.
// ---------------------------------------------------------------------------

typedef __attribute__((ext_vector_type(2))) float v2f;
typedef __attribute__((ext_vector_type(8))) float v8f;

#define IN_CH 128
#define HID   128
#define OUTC  40
#define OUTP  48      // OUTC padded to 3 x 16 WMMA tiles
#define LDA   132     // LDS row stride (floats): 132 % 64 = 4 -> no bank conflicts

// ---------------- small utility kernels ----------------

__global__ void fill_i32(int* __restrict__ p, int v, int n) {
  int i = blockIdx.x * blockDim.x + threadIdx.x;
  if (i < n) p[i] = v;
}

__global__ void copy_i32(const int* __restrict__ a, int* __restrict__ b, int n) {
  int i = blockIdx.x * blockDim.x + threadIdx.x;
  if (i < n) b[i] = a[i];
}

__global__ void deg_count(const long long* __restrict__ dst, int* __restrict__ degI, int E) {
  int e = blockIdx.x * blockDim.x + threadIdx.x;
  if (e < E) atomicAdd(&degI[(int)dst[e]], 1);
}

// repack [K, OUTC] -> [K, OUTP] zero-padded so WMMA N-tiles need no guards
__global__ void pad_w(const float* __restrict__ W, float* __restrict__ Wp, int K) {
  int i = blockIdx.x * blockDim.x + threadIdx.x;
  if (i >= K * OUTP) return;
  int r = i / OUTP, c = i % OUTP;
  Wp[i] = (c < OUTC) ? W[r * OUTC + c] : 0.0f;
}

// ---------------- exclusive scan (3 kernels, LDS Hillis-Steele) ----------------

#define SCAN_B 256
__global__ void scan1(const int* __restrict__ in, int* __restrict__ out,
                      int* __restrict__ blockSums, int n) {
  __shared__ int sh[SCAN_B];
  const int i = blockIdx.x * SCAN_B + threadIdx.x;
  const int v = (i < n) ? in[i] : 0;
  sh[threadIdx.x] = v;
  __syncthreads();
  for (int off = 1; off < SCAN_B; off <<= 1) {
    int t = (threadIdx.x >= off) ? sh[threadIdx.x - off] : 0;
    __syncthreads();
    sh[threadIdx.x] += t;
    __syncthreads();
  }
  if (i < n) out[i] = sh[threadIdx.x] - v;                 // exclusive
  if (threadIdx.x == SCAN_B - 1) blockSums[blockIdx.x] = sh[threadIdx.x];
}

__global__ void scan2(int* __restrict__ blockSums, int nb) {  // single block, nb <= 512
  __shared__ int sh[512];
  const int v = (threadIdx.x < nb) ? blockSums[threadIdx.x] : 0;
  sh[threadIdx.x] = v;
  __syncthreads();
  for (int off = 1; off < 512; off <<= 1) {
    int t = (threadIdx.x >= off) ? sh[threadIdx.x - off] : 0;
    __syncthreads();
    sh[threadIdx.x] += t;
    __syncthreads();
  }
  if (threadIdx.x < nb) blockSums[threadIdx.x] = sh[threadIdx.x] - v;  // exclusive
}

__global__ void scan3(int* __restrict__ out, const int* __restrict__ blockSums, int n) {
  int i = blockIdx.x * SCAN_B + threadIdx.x;
  if (i < n) out[i] += blockSums[blockIdx.x];
}

// bucket edges by destination: srcSorted[start[dst] ...] = src
__global__ void bucket_edges(const long long* __restrict__ src,
                             const long long* __restrict__ dst,
                             int* __restrict__ pos, int* __restrict__ srcSorted, int E) {
  int e = blockIdx.x * blockDim.x + threadIdx.x;
  if (e >= E) return;
  int idx = atomicAdd(&pos[(int)dst[e]], 1);
  srcSorted[idx] = (int)src[e];
}

// ---------------- CSR pull aggregation (no atomics) ----------------

// aggr[node] = mean over in-edges of q[src], 128 ch; one wave/node, float4/lane
__global__ void aggregate128(const int* __restrict__ start, const int* __restrict__ degI,
                             const int* __restrict__ srcS, const float* __restrict__ q,
                             float* __restrict__ aggr, int N) {
  const int node = blockIdx.x * (blockDim.x >> 5) + (threadIdx.x >> 5);
  if (node >= N) return;
  const int lane = threadIdx.x & 31;
  const int s0 = start[node], d = degI[node];
  float4 acc = make_float4(0.f, 0.f, 0.f, 0.f);
  for (int j = 0; j < d; ++j) {
    const int s = srcS[s0 + j];
    const float4 v = ((const float4*)(q + (size_t)s * HID))[lane];
    acc.x += v.x; acc.y += v.y; acc.z += v.z; acc.w += v.w;
  }
  const float inv = 1.0f / fmaxf((float)d, 1.0f);
  acc.x *= inv; acc.y *= inv; acc.z *= inv; acc.w *= inv;
  ((float4*)(aggr + (size_t)node * HID))[lane] = acc;
}

// aggr40[node] = mean over in-edges of p[src], 40 ch
__global__ void aggregate40(const int* __restrict__ start, const int* __restrict__ degI,
                            const int* __restrict__ srcS, const float* __restrict__ p,
                            float* __restrict__ aggr40, int N) {
  const int node = blockIdx.x * (blockDim.x >> 5) + (threadIdx.x >> 5);
  if (node >= N) return;
  const int lane = threadIdx.x & 31;
  const int s0 = start[node], d = degI[node];
  float a0 = 0.f, a1 = 0.f;
  for (int j = 0; j < d; ++j) {
    const float* pr = p + (size_t)srcS[s0 + j] * OUTC;
    a0 += pr[lane];
    if (lane < 8) a1 += pr[32 + lane];
  }
  const float inv = 1.0f / fmaxf((float)d, 1.0f);
  aggr40[(size_t)node * OUTC + lane] = a0 * inv;
  if (lane < 8) aggr40[(size_t)node * OUTC + 32 + lane] = a1 * inv;
}

// ---------------- WMMA GEMM kernels ----------------
// Persistent blocks: B-slice (128x16 per wave) lives in 64 VGPRs for all row
// tiles; A tile (16x128) staged in LDS once per tile, shared by all waves.

__device__ __forceinline__ void load_breg(const float* __restrict__ W, int ldB,
                                          int col, int half2, v2f breg[32]) {
#pragma unroll
  for (int kk = 0; kk < 32; ++kk) {
    const int ka = kk * 4 + half2;
    breg[kk].x = W[(size_t)ka * ldB + col];
    breg[kk].y = W[(size_t)(ka + 1) * ldB + col];
  }
}

__device__ __forceinline__ v8f wmma_tile(const float* __restrict__ arow, int half2,
                                         const v2f breg[32]) {
  v8f acc = {};
#pragma unroll
  for (int kk = 0; kk < 32; ++kk) {
    const int ka = kk * 4 + half2;
    v2f a; a.x = arow[ka]; a.y = arow[ka + 1];    // ds_load_b64 from LDS
    acc = __builtin_amdgcn_wmma_f32_16x16x4_f32(false, a, false, breg[kk],
                                                (short)0, acc, false, false);
  }
  return acc;
}

// q = x @ Wl1   (256 threads = 8 waves -> 128 output cols)
__global__ void gemm_q1(const float* __restrict__ x, const float* __restrict__ W,
                        float* __restrict__ q, int rowTiles) {
  __shared__ float sA[16 * LDA];
  const int lane = threadIdx.x & 31;
  const int wave = threadIdx.x >> 5;
  const int l15 = lane & 15;
  const int half2 = (lane >> 4) << 1;
  const int rOff = (lane >> 4) << 3;
  const int col = wave * 16 + l15;

  v2f breg[32];
  load_breg(W, HID, col, half2, breg);

  for (int tile = blockIdx.x; tile < rowTiles; tile += gridDim.x) {
    const size_t rowBase = (size_t)tile * 16;
    __syncthreads();
    for (int i = threadIdx.x; i < 16 * 32; i += blockDim.x) {
      const int r = i >> 5, c4 = i & 31;
      ((float4*)(sA + r * LDA))[c4] = ((const float4*)(x + (rowBase + r) * IN_CH))[c4];
    }
    __syncthreads();
    v8f acc = wmma_tile(sA + l15 * LDA, half2, breg);
#pragma unroll
    for (int i = 0; i < 8; ++i)
      q[(rowBase + rOff + i) * HID + col] = acc[i];
  }
}

// h = relu(x @ Wr1 + b1 + aggr)          (aggr already mean-scaled)
__global__ void gemm_h1(const float* __restrict__ x, const float* __restrict__ W,
                        const float* __restrict__ b1, const float* __restrict__ aggr,
                        float* __restrict__ h, int rowTiles) {
  __shared__ float sA[16 * LDA];
  const int lane = threadIdx.x & 31;
  const int wave = threadIdx.x >> 5;
  const int l15 = lane & 15;
  const int half2 = (lane >> 4) << 1;
  const int rOff = (lane >> 4) << 3;
  const int col = wave * 16 + l15;

  v2f breg[32];
  load_breg(W, HID, col, half2, breg);
  const float bc = b1[col];

  for (int tile = blockIdx.x; tile < rowTiles; tile += gridDim.x) {
    const size_t rowBase = (size_t)tile * 16;
    __syncthreads();
    for (int i = threadIdx.x; i < 16 * 32; i += blockDim.x) {
      const int r = i >> 5, c4 = i & 31;
      ((float4*)(sA + r * LDA))[c4] = ((const float4*)(x + (rowBase + r) * IN_CH))[c4];
    }
    __syncthreads();
    v8f acc = wmma_tile(sA + l15 * LDA, half2, breg);
#pragma unroll
    for (int i = 0; i < 8; ++i) {
      const size_t row = rowBase + rOff + i;
      const float v = acc[i] + bc + aggr[row * HID + col];
      h[row * HID + col] = fmaxf(v, 0.0f);
    }
  }
}

// p = h @ Wl2p   (96 threads = 3 waves -> 48 padded cols, store cols < 40)
__global__ void gemm_p2(const float* __restrict__ h, const float* __restrict__ Wp,
                        float* __restrict__ p, int rowTiles) {
  __shared__ float sA[16 * LDA];
  const int lane = threadIdx.x & 31;
  const int wave = threadIdx.x >> 5;
  const int l15 = lane & 15;
  const int half2 = (lane >> 4) << 1;
  const int rOff = (lane >> 4) << 3;
  const int col = wave * 16 + l15;

  v2f breg[32];
  load_breg(Wp, OUTP, col, half2, breg);

  for (int tile = blockIdx.x; tile < rowTiles; tile += gridDim.x) {
    const size_t rowBase = (size_t)tile * 16;
    __syncthreads();
    for (int i = threadIdx.x; i < 16 * 32; i += blockDim.x) {
      const int r = i >> 5, c4 = i & 31;
      ((float4*)(sA + r * LDA))[c4] = ((const float4*)(h + (rowBase + r) * HID))[c4];
    }
    __syncthreads();
    v8f acc = wmma_tile(sA + l15 * LDA, half2, breg);
    if (col < OUTC) {
#pragma unroll
      for (int i = 0; i < 8; ++i)
        p[(rowBase + rOff + i) * OUTC + col] = acc[i];
    }
  }
}

// logits = h @ Wr2p + b2 + aggr40        (aggr40 already mean-scaled)
__global__ void gemm_logits2(const float* __restrict__ h, const float* __restrict__ Wp,
                             const float* __restrict__ b2, const float* __restrict__ aggr40,
                             float* __restrict__ logits, int rowTiles) {
  __shared__ float sA[16 * LDA];
  const int lane = threadIdx.x & 31;
  const int wave = threadIdx.x >> 5;
  const int l15 = lane & 15;
  const int half2 = (lane >> 4) << 1;
  const int rOff = (lane >> 4) << 3;
  const int col = wave * 16 + l15;

  v2f breg[32];
  load_breg(Wp, OUTP, col, half2, breg);
  const float bc = (col < OUTC) ? b2[col] : 0.0f;

  for (int tile = blockIdx.x; tile < rowTiles; tile += gridDim.x) {
    const size_t rowBase = (size_t)tile * 16;
    __syncthreads();
    for (int i = threadIdx.x; i < 16 * 32; i += blockDim.x) {
      const int r = i >> 5, c4 = i & 31;
      ((float4*)(sA + r * LDA))[c4] = ((const float4*)(h + (rowBase + r) * HID))[c4];
    }
    __syncthreads();
    v8f acc = wmma_tile(sA + l15 * LDA, half2, breg);
    if (col < OUTC) {
#pragma unroll
      for (int i = 0; i < 8; ++i) {
        const size_t row = rowBase + rOff + i;
        logits[row * OUTC + col] = acc[i] + bc + aggr40[row * OUTC + col];
      }
    }
  }
}

// ---------------- final log-softmax, one thread per row ----------------
__global__ void log_softmax40(const float* __restrict__ logits,
                              float* __restrict__ out, int N) {
  int row = blockIdx.x * blockDim.x + threadIdx.x;
  if (row >= N) return;
  const float* pr = logits + (size_t)row * OUTC;
  float m = pr[0];
#pragma unroll
  for (int c = 1; c < OUTC; ++c) m = fmaxf(m, pr[c]);
  float s = 0.0f;
#pragma unroll
  for (int c = 0; c < OUTC; ++c) s += __expf(pr[c] - m);
  const float ls = m + __logf(s);
#pragma unroll
  for (int c = 0; c < OUTC; ++c) out[(size_t)row * OUTC + c] = pr[c] - ls;
}

// ---------------------------------------------------------------------------

extern "C" void kernel_launch(void* const* d_in, const int* in_sizes, int n_in,
                              void* d_out, int out_size, void* d_ws, size_t ws_size,
                              hipStream_t stream) {
  const float*     x   = (const float*)d_in[0];
  const long long* ei  = (const long long*)d_in[1];   // int64 [2, E]
  const float*     Wl1 = (const float*)d_in[2];
  const float*     b1  = (const float*)d_in[3];
  const float*     Wr1 = (const float*)d_in[4];
  const float*     Wl2 = (const float*)d_in[5];
  const float*     b2  = (const float*)d_in[6];
  const float*     Wr2 = (const float*)d_in[7];
  float*           out = (float*)d_out;

  const int N = in_sizes[0] / IN_CH;     // 100000 (multiple of 16)
  const int E = in_sizes[1] / 2;         // 1600000
  const long long* src = ei;
  const long long* dst = ei + E;

  // ---- workspace layout ----
  int*   degI  = (int*)d_ws;                       // N
  int*   start = degI + N;                         // N
  int*   pos   = start + N;                        // N
  int*   bsums = pos + N;                          // 512
  int*   srcS  = bsums + 512;                      // E
  float* Wl2p  = (float*)(srcS + E);               // 128*48
  float* Wr2p  = Wl2p + (size_t)HID * OUTP;        // 128*48
  float* q     = Wr2p + (size_t)HID * OUTP;        // N*128   (later h)
  float* aggr  = q + (size_t)N * HID;              // N*128   (later p/aggr40/logits)
  float* p      = aggr;                            // N*40  (aggr dead after gemm_h1)
  float* aggr40 = aggr + (size_t)N * OUTC;         // N*40
  float* logits = aggr + 2 * (size_t)N * OUTC;     // N*40

  const int rowTiles   = N / 16;                   // 6250
  const int scanBlocks = (N + SCAN_B - 1) / SCAN_B;
  const int gemmGrid   = (rowTiles < 1536) ? rowTiles : 1536;

  // ---- build CSR (dst-sorted src list) ----
  fill_i32<<<(N + 255) / 256, 256, 0, stream>>>(degI, 0, N);
  deg_count<<<(E + 255) / 256, 256, 0, stream>>>(dst, degI, E);
  scan1<<<scanBlocks, SCAN_B, 0, stream>>>(degI, start, bsums, N);
  scan2<<<1, 512, 0, stream>>>(bsums, scanBlocks);
  scan3<<<scanBlocks, SCAN_B, 0, stream>>>(start, bsums, N);
  copy_i32<<<(N + 255) / 256, 256, 0, stream>>>(start, pos, N);
  bucket_edges<<<(E + 255) / 256, 256, 0, stream>>>(src, dst, pos, srcS, E);

  // ---- padded layer-2 weights ----
  pad_w<<<(HID * OUTP + 255) / 256, 256, 0, stream>>>(Wl2, Wl2p, HID);
  pad_w<<<(HID * OUTP + 255) / 256, 256, 0, stream>>>(Wr2, Wr2p, HID);

  // ---- layer 1 ----
  gemm_q1<<<gemmGrid, 256, 0, stream>>>(x, Wl1, q, rowTiles);
  aggregate128<<<(N * 32 + 255) / 256, 256, 0, stream>>>(start, degI, srcS, q, aggr, N);
  gemm_h1<<<gemmGrid, 256, 0, stream>>>(x, Wr1, b1, aggr, q, rowTiles);  // h -> q buffer

  // ---- layer 2 ----
  gemm_p2<<<gemmGrid, 96, 0, stream>>>(q, Wl2p, p, rowTiles);
  aggregate40<<<(N * 32 + 255) / 256, 256, 0, stream>>>(start, degI, srcS, p, aggr40, N);
  gemm_logits2<<<gemmGrid, 96, 0, stream>>>(q, Wr2p, b2, aggr40, logits, rowTiles);

  // ---- log-softmax ----
  log_softmax40<<<(N + 255) / 256, 256, 0, stream>>>(logits, out, N);
}